// MultiViewLSSFisheye_64415919505618
// MI455X (gfx1250) — compile-verified
//
#include <hip/hip_runtime.h>

// ---------------------------------------------------------------------------
// LSS voxel pooling (MultiViewLSSFisheye): scatter-add 461,824 points x 80
// channels into a (4, 80, 200, 200) BEV grid, scaled by 1/NUM_VIEWS.
// Memory-bound (~200 MB traffic -> ~9 us HBM floor on MI455X). No GEMM
// structure -> WMMA inapplicable (AI ~ 0.2 FLOP/byte); optimize the memory
// system instead:
//  - wave32 ballot compaction of kept points
//  - NT-hinted streaming loads (preserve 192 MB L2 for the 51 MB accumulator)
//  - global_prefetch_b8 pipelining of the next kept point's feature row
//  - f32 atomics resolving in L2 (output fits entirely in L2)
// ---------------------------------------------------------------------------

namespace {

constexpr int  Bb    = 4;
constexpr int  Nn    = 4;
constexpr int  Dd    = 41;
constexpr int  Hh    = 16;
constexpr int  Ww    = 44;
constexpr int  Cc    = 80;
constexpr int  NXg   = 200;
constexpr int  NYg   = 200;
constexpr int  NZg   = 1;
constexpr long long PB = (long long)Nn * Dd * Hh * Ww;   // 115456 pts / batch
constexpr long long P  = (long long)Bb * PB;             // 461824 pts total
constexpr int  PLANE   = NXg * NYg;                      // 40000
constexpr int  OUTN    = Bb * Cc * PLANE;                // 12,800,000 floats

// Defined FIRST so the disasm snippet shows this kernel's codegen
// (prefetch / NT hints / atomic selection).
__global__ __launch_bounds__(256) void voxel_scatter(
    const float* __restrict__ geom,          // (P, 3) camera-frame points
    const unsigned char* __restrict__ mask,  // (P,) bool (1-byte jax bool_)
    const float* __restrict__ x,             // (P, 80) features
    float* __restrict__ out)                 // (4, 80, 200, 200)
{
  const int lane = threadIdx.x & 31;
  const long long pblk = (long long)blockIdx.x * 256;
  const long long p = pblk + threadIdx.x;      // this lane's point
  const long long p0 = p - lane;               // wave's first point

  // PB % 256 == 0 -> a block never straddles a batch: b is block-uniform.
  const int b = (int)(pblk / PB);

  // --- per-lane point evaluation (coalesced, streamed once: NT) -----------
  const float g0 = __builtin_nontemporal_load(&geom[p * 3 + 0]);
  const float g1 = __builtin_nontemporal_load(&geom[p * 3 + 1]);
  const float g2 = __builtin_nontemporal_load(&geom[p * 3 + 2]);
  // idx = trunc((g - lo)/dx); lo = bx - dx/2 = {-50,-50,-10} is exact in f32.
  // C-cast truncation toward zero == jnp .astype(int32). /0.5 -> exact *2,
  // /20 stays a true division (matches JAX rounding).
  const int ix = (int)((g0 + 50.0f) / 0.5f);
  const int iy = (int)((g1 + 50.0f) / 0.5f);
  const int iz = (int)((g2 + 10.0f) / 20.0f);
  const unsigned char mv = __builtin_nontemporal_load(&mask[p]);

  const bool kept = (ix >= 0) & (ix < NXg) & (iy >= 0) & (iy < NYg) &
                    (iz >= 0) & (iz < NZg) & (mv != 0);

  // out layout (B, NZ*C, NX, NY), NZ=1: base = b*C*PLANE + ix*NY + iy,
  // channel c lives at +c*PLANE.
  const int obase = b * (Cc * PLANE) + ix * NYg + iy;

  // --- wave32 compaction: full wave serves each surviving point -----------
  unsigned int ball = __builtin_amdgcn_ballot_w32(kept);
  while (ball) {
    const int j = __builtin_ctz(ball);
    ball &= ball - 1u;

    // Prefetch the NEXT kept point's 320B feature row (3 cachelines) as
    // early as possible -> global_prefetch_b8 overlaps this point's loads
    // and atomics with the next point's HBM fetch.
    if (ball) {
      const int jn = __builtin_ctz(ball);
      const char* nxt = (const char*)(x + (p0 + jn) * (long long)Cc);
      if (lane < 3) __builtin_prefetch(nxt + lane * 128, 0, 1);
    }

    const int ob = __builtin_amdgcn_ds_bpermute(j << 2, obase);
    const float* __restrict__ xp = x + (p0 + j) * (long long)Cc;

    // lane c handles channels c, c+32, c+64 -> coalesced 128B NT loads.
    const float v0 = __builtin_nontemporal_load(&xp[lane]);
    const float v1 = __builtin_nontemporal_load(&xp[lane + 32]);
    float v2 = 0.f;
    if (lane < 16) v2 = __builtin_nontemporal_load(&xp[lane + 64]);

    // Fold the /NUM_VIEWS into the scattered value; atomics resolve in L2
    // (51.2 MB accumulator << 192 MB L2, kept resident by NT streams).
    atomicAdd(&out[ob + lane * PLANE],        v0 * 0.25f);
    atomicAdd(&out[ob + (lane + 32) * PLANE], v1 * 0.25f);
    if (lane < 16) atomicAdd(&out[ob + (lane + 64) * PLANE], v2 * 0.25f);
  }
}

__global__ __launch_bounds__(256) void zero_kernel(float4* __restrict__ o) {
  const int i = blockIdx.x * 256 + threadIdx.x;
  o[i] = make_float4(0.f, 0.f, 0.f, 0.f);
}

} // namespace

extern "C" void kernel_launch(void* const* d_in, const int* in_sizes, int n_in,
                              void* d_out, int out_size, void* d_ws, size_t ws_size,
                              hipStream_t stream) {
  (void)in_sizes; (void)n_in; (void)d_ws; (void)ws_size; (void)out_size;

  const float*         geom = (const float*)d_in[0];          // (B,N,D,H,W,3) f32
  const unsigned char* mask = (const unsigned char*)d_in[1];  // (B,N,D,H,W) bool
  const float*         x    = (const float*)d_in[2];          // (B,N,D,H,W,C) f32
  float*               out  = (float*)d_out;                  // (B, C, NX, NY) f32

  // 1) zero the accumulator (harness poisons d_out; atomics need zeros)
  {
    const int n4 = OUTN / 4;                  // 3,200,000 float4
    zero_kernel<<<n4 / 256, 256, 0, stream>>>((float4*)out);
  }

  // 2) scatter-add; 461,824 points / 256 per block = 1804 blocks exactly
  {
    const int blocks = (int)(P / 256);        // PB % 256 == 0 -> exact
    voxel_scatter<<<blocks, 256, 0, stream>>>(geom, mask, x, out);
  }
}